// MultiBoxLoss_91173565759797
// MI455X (gfx1250) — compile-verified
//
#include <hip/hip_runtime.h>
#include <hip/hip_bf16.h>
#include <cstdint>

// MultiBoxLoss (SSD) on MI455X / gfx1250, wave32.
// Roofline: single 88MB read of conf_pred dominates (~4us @ 23.3 TB/s HBM).
// conf_pred tiles are staged global->LDS with GLOBAL_LOAD_ASYNC_TO_LDS_B128
// (ASYNCcnt path, coalesced b128, no VGPR round trip); WMMA f32 16x16x4 does
// the f32 wave reductions on the matrix pipe (co-executes with VALU exp/log);
// radix-select replaces the double argsort for hard-negative mining
// (sum of top-k is tie-order invariant).

#define B_ 32
#define P_ 32768
#define T_ 64
#define C_ 21

typedef float v2f __attribute__((ext_vector_type(2)));
typedef float v8f __attribute__((ext_vector_type(8)));

// ---- CDNA5 async global->LDS 16B copy (per active lane), ASYNCcnt-tracked.
__device__ __forceinline__ void async_copy_b128(unsigned lds_off,
                                                const void* gptr) {
  asm volatile("global_load_async_to_lds_b128 %0, %1, off"
               :: "v"(lds_off),
                  "v"((unsigned long long)(uintptr_t)gptr)
               : "memory");
}

__device__ __forceinline__ void wait_async0() {
#if __has_builtin(__builtin_amdgcn_s_wait_asynccnt)
  __builtin_amdgcn_s_wait_asynccnt(0);
#else
  asm volatile("s_wait_asynccnt 0x0" ::: "memory");
#endif
}

// ---------------------------------------------------------------- kernel 0
__global__ void k_init(int* __restrict__ num_pos,
                       float* __restrict__ loc_sum,
                       float* __restrict__ posce_sum) {
  if (threadIdx.x < B_) num_pos[threadIdx.x] = 0;
  if (threadIdx.x == 0) { *loc_sum = 0.f; *posce_sum = 0.f; }
}

// ---------------------------------------------------------------- kernel 1
// Per-prior best truth: best_ovr[b,p] = max_t IoU, best_idx[b,p] = first argmax.
__global__ void k_best_truth(const float* __restrict__ priors,
                             const float* __restrict__ tboxes,
                             float* __restrict__ best_ovr,
                             int* __restrict__ best_idx) {
  __shared__ float4 s_t[T_];
  __shared__ float  s_area[T_];
  const int chunks = P_ / 256;
  const int b = blockIdx.x / chunks;
  const int p = (blockIdx.x % chunks) * 256 + threadIdx.x;
  if (threadIdx.x < T_) {
    float4 t = ((const float4*)tboxes)[(size_t)b * T_ + threadIdx.x];
    s_t[threadIdx.x] = t;
    s_area[threadIdx.x] = (t.z - t.x) * (t.w - t.y);
  }
  __syncthreads();
  const float4 pv = ((const float4*)priors)[p];
  const float px1 = pv.x - 0.5f * pv.z, py1 = pv.y - 0.5f * pv.w;
  const float px2 = pv.x + 0.5f * pv.z, py2 = pv.y + 0.5f * pv.w;
  const float area_b = pv.z * pv.w;
  float best = -1.0f; int bi = 0;
#pragma unroll 4
  for (int t = 0; t < T_; ++t) {
    float4 tb = s_t[t];
    float ix = fmaxf(fminf(px2, tb.z) - fmaxf(px1, tb.x), 0.f);
    float iy = fmaxf(fminf(py2, tb.w) - fmaxf(py1, tb.y), 0.f);
    float inter = ix * iy;
    float iou = inter / (s_area[t] + area_b - inter);
    if (iou > best) { best = iou; bi = t; }   // strict '>' keeps first argmax
  }
  best_ovr[(size_t)b * P_ + p] = best;
  best_idx[(size_t)b * P_ + p] = bi;
}

// ---------------------------------------------------------------- kernel 2
// Per-truth best prior (argmax over P, first-occurrence tie-break).
__global__ void k_best_prior(const float* __restrict__ priors,
                             const float* __restrict__ tboxes,
                             int* __restrict__ best_prior_idx) {
  const int b = blockIdx.x / T_;
  const int t = blockIdx.x % T_;
  const float4 tb = ((const float4*)tboxes)[(size_t)b * T_ + t];
  const float area_a = (tb.z - tb.x) * (tb.w - tb.y);
  float best = -1.f; int bp = 0;
  for (int p = threadIdx.x; p < P_; p += 256) {
    float4 pv = ((const float4*)priors)[p];
    float ix = fmaxf(fminf(pv.x + 0.5f * pv.z, tb.z) -
                     fmaxf(pv.x - 0.5f * pv.z, tb.x), 0.f);
    float iy = fmaxf(fminf(pv.y + 0.5f * pv.w, tb.w) -
                     fmaxf(pv.y - 0.5f * pv.w, tb.y), 0.f);
    float inter = ix * iy;
    float iou = inter / (area_a + pv.z * pv.w - inter);
    if (iou > best) { best = iou; bp = p; }   // per-thread p strictly increases
  }
  __shared__ float sv[256];
  __shared__ int   sp[256];
  sv[threadIdx.x] = best; sp[threadIdx.x] = bp;
  __syncthreads();
  for (int s = 128; s > 0; s >>= 1) {
    if ((int)threadIdx.x < s) {
      float ov = sv[threadIdx.x + s]; int op = sp[threadIdx.x + s];
      if (ov > sv[threadIdx.x] ||
          (ov == sv[threadIdx.x] && op < sp[threadIdx.x])) {
        sv[threadIdx.x] = ov; sp[threadIdx.x] = op;
      }
    }
    __syncthreads();
  }
  if (threadIdx.x == 0) best_prior_idx[blockIdx.x] = sp[0];
}

// ---------------------------------------------------------------- kernel 3
// Force-match: serial over t so "last truth wins" on duplicate priors.
__global__ void k_force(const int* __restrict__ bpi,
                        float* __restrict__ best_ovr,
                        int* __restrict__ best_idx) {
  const int b = threadIdx.x;
  if (b >= B_) return;
  for (int t = 0; t < T_; ++t) {
    int p = bpi[b * T_ + t];
    best_ovr[(size_t)b * P_ + p] = 2.0f;
    best_idx[(size_t)b * P_ + p] = t;
  }
}

// ---------------------------------------------------------------- kernel 4
// Fused streaming pass: conf tile async-staged to LDS, CE (log-softmax),
// smooth-L1 loc loss, pos count. WMMA f32 16x16x4 with B=ones performs the
// f32 wave reduction on the matrix pipe; D row-sums collapse via sum over
// 8 acc VGPRs + shfl_xor(16).
__global__ void k_main(const float* __restrict__ loc_pred,
                       const float* __restrict__ conf_pred,
                       const float* __restrict__ priors,
                       const float* __restrict__ tboxes,
                       const int* __restrict__ labels,
                       const float* __restrict__ best_ovr,
                       const int* __restrict__ best_idx,
                       float* __restrict__ neg_conf,
                       int* __restrict__ num_pos,
                       float* __restrict__ loc_sum,
                       float* __restrict__ posce_sum) {
  __shared__ float s_conf[256 * C_];               // 21504 B, 16B aligned tile
  const int chunks = P_ / 256;
  const int b = blockIdx.x / chunks;
  const int p0 = (blockIdx.x % chunks) * 256;
  const int p = p0 + threadIdx.x;
  const size_t bp = (size_t)b * P_ + p;

  // ---- async global->LDS staging of this block's conf tile (coalesced b128)
  {
    const float* gbase = conf_pred + ((size_t)b * P_ + p0) * C_;
    const unsigned lds0 = (unsigned)(uintptr_t)&s_conf[0];
    const int nvec = (256 * C_) / 4;               // 1344 x 16B
    for (int i = threadIdx.x; i < nvec; i += 256)
      async_copy_b128(lds0 + 16u * (unsigned)i, gbase + 4 * i);
    wait_async0();
  }

  const int   bt  = best_idx[bp];
  const float ovr = best_ovr[bp];
  const int   lab = labels[b * T_ + bt];
  const int   conf_t = (ovr < 0.5f) ? 0 : lab;
  const bool  pos = conf_t > 0;

  __syncthreads();                                 // all waves' tiles landed

  // ---- cross entropy over C=21 classes (stride-21 LDS: bank-conflict free)
  const float* xs = &s_conf[threadIdx.x * C_];
  float x[C_];
  float m = -3.4e38f;
#pragma unroll
  for (int c = 0; c < C_; ++c) { x[c] = xs[c]; m = fmaxf(m, x[c]); }
  float s = 0.f;
#pragma unroll
  for (int c = 0; c < C_; ++c) s += expf(x[c] - m);
  const float ce = (m + logf(s)) - x[conf_t];

  neg_conf[bp] = pos ? 0.f : ce;
  const float posce = pos ? ce : 0.f;

  // ---- smooth-L1 localization loss on positives
  float l = 0.f;
  if (pos) {
    const float4 tb = ((const float4*)tboxes)[(size_t)b * T_ + bt];
    const float4 pv = ((const float4*)priors)[p];
    const float4 lp = ((const float4*)loc_pred)[bp];
    float g[4], q[4];
    g[0] = ((tb.x + tb.z) * 0.5f - pv.x) / (0.1f * pv.z);
    g[1] = ((tb.y + tb.w) * 0.5f - pv.y) / (0.1f * pv.w);
    g[2] = logf((tb.z - tb.x) / pv.z) / 0.2f;
    g[3] = logf((tb.w - tb.y) / pv.w) / 0.2f;
    q[0] = lp.x; q[1] = lp.y; q[2] = lp.z; q[3] = lp.w;
#pragma unroll
    for (int j = 0; j < 4; ++j) {
      float d = fabsf(q[j] - g[j]);
      l += (d < 1.f) ? 0.5f * d * d : d - 0.5f;
    }
  }

  // ---- wave reduction via matrix pipe (EXEC is all-ones here: full blocks,
  // branches above have reconverged). Each lane feeds (M=lane%16, K=lane<16?0:2).
  v2f onesB = {1.f, 1.f};
  v2f aL = {l, 0.f};
  v2f aC = {posce, 0.f};
  v8f accL = {};
  v8f accC = {};
  accL = __builtin_amdgcn_wmma_f32_16x16x4_f32(false, aL, false, onesB,
                                               (short)0, accL, false, false);
  accC = __builtin_amdgcn_wmma_f32_16x16x4_f32(false, aC, false, onesB,
                                               (short)0, accC, false, false);
  float sl = 0.f, sc = 0.f;
#pragma unroll
  for (int j = 0; j < 8; ++j) { sl += accL[j]; sc += accC[j]; }
  sl += __shfl_xor(sl, 16, 32);   // rows 0..7 + rows 8..15
  sc += __shfl_xor(sc, 16, 32);

  unsigned long long mpos = __ballot(pos);
  if ((threadIdx.x & 31) == 0) {
    atomicAdd(loc_sum, sl);
    atomicAdd(posce_sum, sc);
    atomicAdd(&num_pos[b], (int)__popcll(mpos));
  }
}

// ---------------------------------------------------------------- kernel 5
// Per-batch sum of top-k neg_conf via exact 8-bit radix select on float bits
// (values >= 0 so uint ordering == float ordering). L2-resident data.
__global__ void k_topk(const float* __restrict__ neg_conf,
                       const int* __restrict__ num_pos,
                       float* __restrict__ topk_sum) {
  const int b = blockIdx.x;
  const float* v = neg_conf + (size_t)b * P_;
  int k = 3 * num_pos[b];
  if (k > P_ - 1) k = P_ - 1;

  __shared__ unsigned hist[256];
  __shared__ int s_rem, s_bin;
  __shared__ float ss[256];

  if (k <= 0) {
    if (threadIdx.x == 0) topk_sum[b] = 0.f;
    return;
  }

  unsigned pref = 0;
  int rem = k;
  for (int pass = 0; pass < 4; ++pass) {
    const int shift = 24 - 8 * pass;
    const unsigned mask = pass ? (0xFFFFFFFFu << (shift + 8)) : 0u;
    hist[threadIdx.x] = 0;
    __syncthreads();
    for (int i = threadIdx.x; i < P_; i += 256) {
      unsigned u = __float_as_uint(v[i]);
      if ((u & mask) == pref) atomicAdd(&hist[(u >> shift) & 0xFFu], 1u);
    }
    __syncthreads();
    if (threadIdx.x == 0) {
      int r = rem, binSel = 0;
      for (int bin = 255; bin >= 0; --bin) {
        int c = (int)hist[bin];
        if (r <= c) { binSel = bin; break; }
        r -= c;
      }
      s_rem = r; s_bin = binSel;
    }
    __syncthreads();
    rem = s_rem;
    pref |= ((unsigned)s_bin) << shift;
    __syncthreads();
  }
  // pref = k-th largest value's bits; rem = copies of it inside the top-k.
  const float kv = __uint_as_float(pref);
  float acc = 0.f;
  for (int i = threadIdx.x; i < P_; i += 256) {
    float x = v[i];
    if (__float_as_uint(x) > pref) acc += x;
  }
  ss[threadIdx.x] = acc;
  __syncthreads();
  for (int s = 128; s > 0; s >>= 1) {
    if ((int)threadIdx.x < s) ss[threadIdx.x] += ss[threadIdx.x + s];
    __syncthreads();
  }
  if (threadIdx.x == 0) topk_sum[b] = ss[0] + (float)rem * kv;
}

// ---------------------------------------------------------------- kernel 6
__global__ void k_final(const int* __restrict__ num_pos,
                        const float* __restrict__ topk_sum,
                        const float* __restrict__ loc_sum,
                        const float* __restrict__ posce_sum,
                        float* __restrict__ out) {
  if (threadIdx.x == 0) {
    int N = 0;
    float conf = *posce_sum;
    for (int b = 0; b < B_; ++b) { N += num_pos[b]; conf += topk_sum[b]; }
    const float fN = (float)N;
    out[0] = *loc_sum / fN;
    out[1] = conf / fN;
  }
}

// ----------------------------------------------------------------
extern "C" void kernel_launch(void* const* d_in, const int* in_sizes, int n_in,
                              void* d_out, int out_size, void* d_ws, size_t ws_size,
                              hipStream_t stream) {
  (void)in_sizes; (void)n_in; (void)out_size; (void)ws_size;
  const float* loc_pred  = (const float*)d_in[0];
  const float* conf_pred = (const float*)d_in[1];
  const float* priors    = (const float*)d_in[2];
  const float* tboxes    = (const float*)d_in[3];
  const int*   labels    = (const int*)d_in[4];
  float* out = (float*)d_out;

  char* ws = (char*)d_ws;
  const size_t BP = (size_t)B_ * P_;                 // 1,048,576
  float* best_ovr = (float*)(ws);                    // BP f32
  int*   best_idx = (int*)  (ws + 4 * BP);           // BP i32
  float* neg_conf = (float*)(ws + 8 * BP);           // BP f32
  int*   bpi      = (int*)  (ws + 12 * BP);          // B*T i32
  int*   num_pos  = (int*)  (ws + 12 * BP + 4 * B_ * T_);
  float* topk     = (float*)(ws + 12 * BP + 4 * B_ * T_ + 4 * B_);
  float* loc_sum  = (float*)(ws + 12 * BP + 4 * B_ * T_ + 8 * B_);
  float* posce    = loc_sum + 1;

  const dim3 blk(256);
  k_init      <<<1, 64, 0, stream>>>(num_pos, loc_sum, posce);
  k_best_truth<<<B_ * (P_ / 256), blk, 0, stream>>>(priors, tboxes, best_ovr, best_idx);
  k_best_prior<<<B_ * T_, blk, 0, stream>>>(priors, tboxes, bpi);
  k_force     <<<1, 32, 0, stream>>>(bpi, best_ovr, best_idx);
  k_main      <<<B_ * (P_ / 256), blk, 0, stream>>>(loc_pred, conf_pred, priors, tboxes,
                                                    labels, best_ovr, best_idx,
                                                    neg_conf, num_pos, loc_sum, posce);
  k_topk      <<<B_, blk, 0, stream>>>(neg_conf, num_pos, topk);
  k_final     <<<1, 32, 0, stream>>>(num_pos, topk, loc_sum, posce, out);
}